// CapsuleNetwork_36069135352418
// MI455X (gfx1250) — compile-verified
//
#include <hip/hip_runtime.h>

typedef __attribute__((ext_vector_type(2))) float v2f;
typedef __attribute__((ext_vector_type(8))) float v8f;
typedef __attribute__((ext_vector_type(4))) unsigned int u32x4;
typedef __attribute__((ext_vector_type(8))) unsigned int u32x8;

#define S_LEN 50
#define E_DIM 64
#define I_NUM 4
#define BT 16                          // batch tile = WMMA M
#define THREADS 512                    // 16 wave32s
#define LDS_HAT (S_LEN * BT * E_DIM)   // 51200 floats = 204.8 KB
#define LDS_CW  (BT * S_LEN)           // 800 floats
#define LDS_MSK (BT * S_LEN)           // 800 ints (TDM destination)
#define LDS_BYTES ((LDS_HAT + LDS_CW + LDS_MSK) * 4)

__global__ __launch_bounds__(THREADS, 1)
void capsule_routing_kernel(const float* __restrict__ item_eb,
                            const int*   __restrict__ mask,
                            const float* __restrict__ w,
                            float* __restrict__ out)
{
    extern __shared__ float smem[];
    float* hat  = smem;                         // [S_LEN][BT][E_DIM]
    float* cwb  = smem + LDS_HAT;               // [BT][S_LEN] routing logits
    int*   smask = (int*)(smem + LDS_HAT + LDS_CW); // [BT][S_LEN] mask tile

    const int tid  = threadIdx.x;
    const int lane = tid & 31;
    const int wv   = tid >> 5;                  // wave id 0..15
    const int l16  = lane & 15;
    const int hi   = lane >> 4;                 // 0 or 1 (half-wave)

    const int interest = blockIdx.x & 3;
    const int b0 = (blockIdx.x >> 2) * BT;

    // ---- TDM: DMA the (16 x 50) i32 mask tile into LDS, overlapped with GEMM.
    // D# group0: count=1, lds_addr, global_addr, type=2.  group1: data_size=4B,
    // tensor 50x16, tile 50x16, dim0 stride 50.  (ISA 8.3/8.4; 2D -> groups 2,3 NULL)
    if (wv == 0) {
        const unsigned long long ga =
            (unsigned long long)(uintptr_t)(mask + (size_t)b0 * S_LEN);
        const unsigned lds_addr = (unsigned)(uintptr_t)(void*)smask;
        u32x4 g0 = { 1u,                               // count=1, user mode
                     lds_addr,                         // bits 63:32  lds_addr
                     (unsigned)ga,                     // bits 95:64  global_addr lo
                     (unsigned)((ga >> 32) & 0x1FFFFFFu) | 0x80000000u }; // ga hi | type=2
        u32x8 g1 = { (2u << 16),                       // data_size = 4B
                     ((unsigned)S_LEN << 16),          // tensor_dim0 = 50
                     ((unsigned)BT << 16),             // tensor_dim1 = 16
                     ((unsigned)S_LEN << 16),          // tile_dim0 = 50
                     (unsigned)BT,                     // tile_dim1 = 16, tile_dim2 = 0
                     (unsigned)S_LEN,                  // tensor_dim0_stride = 50
                     0u, 0u };                         // tensor_dim1_stride = 0
        asm volatile("tensor_load_to_lds %0, %1" : : "s"(g0), "s"(g1) : "memory");
    }

    // ---------------- Phase 1: hat via V_WMMA_F32_16X16X4_F32 ----------------
    // hat[s][m][e] = sum_k item[b0+m, s, k] * w[s, interest*64+e, k]
    // One wave per s: A fragment loaded once, reused by 4 N-tile accumulators.
    for (int s = wv; s < S_LEN; s += 16) {
        const float* aptr  = item_eb + (((size_t)(b0 + l16)) * S_LEN + s) * E_DIM + 2 * hi;
        const float* bbase = w + (((size_t)s) * (I_NUM * E_DIM) + (interest * E_DIM + l16)) * E_DIM + 2 * hi;
        __builtin_prefetch(bbase, 0, 1);        // global_prefetch_b8 (L2-resident w)

        v8f acc0 = {0.f,0.f,0.f,0.f,0.f,0.f,0.f,0.f};
        v8f acc1 = acc0, acc2 = acc0, acc3 = acc0;
        #pragma unroll
        for (int kk = 0; kk < 16; ++kk) {
            v2f a  = *(const v2f*)(aptr + kk * 4);
            v2f b0v = *(const v2f*)(bbase +  0 * E_DIM + kk * 4);
            v2f b1v = *(const v2f*)(bbase + 16 * E_DIM + kk * 4);
            v2f b2v = *(const v2f*)(bbase + 32 * E_DIM + kk * 4);
            v2f b3v = *(const v2f*)(bbase + 48 * E_DIM + kk * 4);
            acc0 = __builtin_amdgcn_wmma_f32_16x16x4_f32(false, a, false, b0v, (short)0, acc0, false, false);
            acc1 = __builtin_amdgcn_wmma_f32_16x16x4_f32(false, a, false, b1v, (short)0, acc1, false, false);
            acc2 = __builtin_amdgcn_wmma_f32_16x16x4_f32(false, a, false, b2v, (short)0, acc2, false, false);
            acc3 = __builtin_amdgcn_wmma_f32_16x16x4_f32(false, a, false, b3v, (short)0, acc3, false, false);
        }
        // D layout: VGPR v holds row M = v + 8*hi, col N = n0 + l16
        #pragma unroll
        for (int v = 0; v < 8; ++v) {
            const int m = v + 8 * hi;
            float* hrow = hat + (s * BT + m) * E_DIM + l16;
            hrow[ 0] = acc0[v];
            hrow[16] = acc1[v];
            hrow[32] = acc2[v];
            hrow[48] = acc3[v];
        }
    }
    for (int p = tid; p < LDS_CW; p += THREADS) cwb[p] = 0.f;
    if (wv == 0) __builtin_amdgcn_s_wait_tensorcnt(0);   // mask tile landed in LDS
    __syncthreads();

    // ---------------- Phase 2: dynamic routing; wave wv owns batch row b0+wv ----
    const int row = b0 + wv;
    float* cw = cwb + wv * S_LEN;
    const float* hrow = hat + wv * E_DIM;               // hrow[s*BT*E + e]
    const int* mrow = smask + wv * S_LEN;               // mask tile in LDS (via TDM)

    float cap0 = 0.f, cap1 = 0.f;                       // lane holds e = lane, lane+32
    #pragma unroll 1
    for (int it = 0; it < 3; ++it) {
        // softmax over S (replicated across lanes; cw broadcast from LDS)
        float mx = -3.0e38f;
        for (int s = 0; s < S_LEN; ++s) mx = fmaxf(mx, cw[s]);
        float sum = 0.f;
        for (int s = 0; s < S_LEN; ++s) sum += __expf(cw[s] - mx);
        const float inv = 1.f / sum;

        cap0 = 0.f; cap1 = 0.f;
        for (int s = 0; s < S_LEN; ++s) {
            const float sv = (mrow[s] == 0) ? 0.f : __expf(cw[s] - mx) * inv;
            cap0 = fmaf(sv, hrow[s * (BT * E_DIM) + lane],      cap0);
            cap1 = fmaf(sv, hrow[s * (BT * E_DIM) + lane + 32], cap1);
        }
        // squash: ||cap||^2 via wave32 butterfly all-reduce
        float nrm = cap0 * cap0 + cap1 * cap1;
        #pragma unroll
        for (int off = 16; off > 0; off >>= 1) nrm += __shfl_xor(nrm, off, 32);
        const float scale = (nrm / (1.f + nrm)) * rsqrtf(nrm + 1e-9f);
        cap0 *= scale; cap1 *= scale;

        if (it < 2) {
            // logit update: delta[s] = hat[s,:] . cap  (allreduce over lanes)
            for (int s = 0; s < S_LEN; ++s) {
                float p = cap0 * hrow[s * (BT * E_DIM) + lane]
                        + cap1 * hrow[s * (BT * E_DIM) + lane + 32];
                #pragma unroll
                for (int off = 16; off > 0; off >>= 1) p += __shfl_xor(p, off, 32);
                if (lane == 0) cw[s] += p;
            }
        }
    }

    float* orow = out + (((size_t)row) * I_NUM + interest) * E_DIM;
    orow[lane]      = cap0;
    orow[lane + 32] = cap1;
}

extern "C" void kernel_launch(void* const* d_in, const int* in_sizes, int n_in,
                              void* d_out, int out_size, void* d_ws, size_t ws_size,
                              hipStream_t stream) {
    const float* item_eb = (const float*)d_in[0];   // (B, 50, 64) f32
    const int*   mask    = (const int*)d_in[1];     // (B, 50) i32
    const float* w       = (const float*)d_in[2];   // (1, 50, 256, 64) f32
    float*       out     = (float*)d_out;           // (B, 4, 64) f32

    const int B = in_sizes[1] / S_LEN;

    (void)hipFuncSetAttribute((const void*)capsule_routing_kernel,
                              hipFuncAttributeMaxDynamicSharedMemorySize, LDS_BYTES);

    dim3 grid((B / BT) * I_NUM);
    capsule_routing_kernel<<<grid, THREADS, LDS_BYTES, stream>>>(item_eb, mask, w, out);
}